// PredictYolov1_1_2542620639985
// MI455X (gfx1250) — compile-verified
//
#include <hip/hip_runtime.h>
#include <hip/hip_bf16.h>

// ---------------------------------------------------------------------------
// YOLOv1 post-process (decode + stable sort + greedy label/batch NMS)
// MI455X (gfx1250, wave32). Latency-bound problem (<1 MB total traffic at
// 23.3 TB/s). Strategy: keep all NMS state resident in one WGP's 320KB LDS,
// stage sorted boxes with async global->LDS B128 copies (ASYNCcnt path).
// ---------------------------------------------------------------------------

#define NB      2
#define NC      20
#define NG      7
#define BATCH   64
#define CELLS   (BATCH * NG * NG)   // 3136
#define NCAND   (CELLS * NB)        // 6272 candidates
#define NPAD    8192                // next pow2 for bitonic sort
#define THR_CONF 0.5f
#define THR_NMS  0.3f

// d_out layout (floats), reference tuple order:
// ids_batch[N], boxes[N*4], labels[N], scores[N], keep[N]  -> 50176 total
#define OFF_IDS  0
#define OFF_BOX  (NCAND)
#define OFF_LAB  (NCAND * 5)
#define OFF_SCO  (NCAND * 6)
#define OFF_KEEP (NCAND * 7)

typedef int vec4i __attribute__((vector_size(16)));
typedef __attribute__((address_space(1))) vec4i as1_v4i;   // global
typedef __attribute__((address_space(3))) vec4i as3_v4i;   // LDS

// Async global->LDS 16B copy (per-lane). Tracked by ASYNCcnt.
__device__ __forceinline__ void async_b128_g2l(const float* g, float* l) {
#if defined(__AMDGCN__) && __has_builtin(__builtin_amdgcn_global_load_async_to_lds_b128)
  __builtin_amdgcn_global_load_async_to_lds_b128((as1_v4i*)g, (as3_v4i*)l, 0, 0);
#else
  unsigned lds_off = (unsigned)(unsigned long long)(as3_v4i*)l;
  asm volatile("global_load_async_to_lds_b128 %0, %1, off"
               :: "v"(lds_off), "v"(g) : "memory");
#endif
}

__device__ __forceinline__ void wait_async0() {
#if defined(__AMDGCN__) && __has_builtin(__builtin_amdgcn_s_wait_asynccnt)
  __builtin_amdgcn_s_wait_asynccnt(0);
#else
  asm volatile("s_wait_asynccnt 0" ::: "memory");
#endif
}

// ---------------------------------------------------------------------------
// Kernel 1: decode. One thread per grid cell (3136 threads total).
// Produces per-candidate: stable 64-bit sort key, box, label, bid, score, valid.
// ---------------------------------------------------------------------------
__global__ void yolo_decode(const float* __restrict__ p,
                            unsigned long long* __restrict__ key,
                            float* __restrict__ box,
                            float* __restrict__ label,
                            float* __restrict__ bid,
                            float* __restrict__ score,
                            float* __restrict__ validf) {
  int cell = blockIdx.x * blockDim.x + threadIdx.x;
  if (cell >= CELLS) return;
  int c = cell % NG;
  int r = (cell / NG) % NG;
  int b = cell / (NG * NG);
  const float* q = p + (long)cell * (NB * (5 + NC));

  float sc[NB];
  int   cls[NB];
#pragma unroll
  for (int e = 0; e < NB; ++e) {
    const float* ep = q + e * (5 + NC);
    float conf = ep[4];
    float mp = ep[5];
    int   mc = 0;
#pragma unroll
    for (int k = 1; k < NC; ++k) {
      float v = ep[5 + k];
      if (v > mp) { mp = v; mc = k; }   // first-max, matches jnp.argmax
    }
    cls[e] = mc;
    sc[e]  = conf * mp;                  // == max(prob*conf) for conf>=0
  }
  int best = (sc[1] > sc[0]) ? 1 : 0;    // argmax over bbox dim (first on tie)
  const float* bp = q + best * (5 + NC);
  float lab = (float)(cls[best] + 1);
  // offxy2xy: rows index -> x, cols index -> y (exactly as reference)
  float x = (bp[0] + (float)r) / (float)NG;
  float y = (bp[1] + (float)c) / (float)NG;
  float hw = bp[2] * 0.5f, hh = bp[3] * 0.5f;
  float bl = x - hw, bt = y - hh, br = x + hw, bb = y + hh;

#pragma unroll
  for (int e = 0; e < NB; ++e) {
    int n = cell * NB + e;
    float s = sc[e];
    bool  valid = s > THR_CONF;
    box[4 * n + 0] = bl;
    box[4 * n + 1] = bt;
    box[4 * n + 2] = br;
    box[4 * n + 3] = bb;
    label[n]  = lab;
    bid[n]    = (float)b;
    score[n]  = s;
    validf[n] = valid ? 1.0f : 0.0f;
    // stable ascending key for where(valid, -score, +inf): ordered-float in
    // high 32 bits, original index as tiebreak in low 32 bits.
    float sv = valid ? -s : __builtin_inff();
    unsigned u = __float_as_uint(sv);
    u = (u & 0x80000000u) ? ~u : (u | 0x80000000u);
    key[n] = (((unsigned long long)u) << 32) | (unsigned)n;
  }
}

// ---------------------------------------------------------------------------
// Kernel 2: single-workgroup bitonic sort of 8192 packed keys in 64KB LDS,
// then gather sorted ids/boxes/labels/scores directly into d_out.
// ---------------------------------------------------------------------------
__global__ __launch_bounds__(1024) void yolo_sort(
    const unsigned long long* __restrict__ key,
    const float* __restrict__ box, const float* __restrict__ label,
    const float* __restrict__ bid, const float* __restrict__ score,
    const float* __restrict__ validf,
    float* __restrict__ out, float* __restrict__ valid_sorted) {
  __shared__ unsigned long long sk[NPAD];   // 64 KB
  int tid = threadIdx.x;
  for (int i = tid; i < NPAD; i += 1024)
    sk[i] = (i < NCAND) ? key[i] : 0xFFFFFFFFFFFFFFFFull;
  __syncthreads();

  for (int k = 2; k <= NPAD; k <<= 1) {
    for (int j = k >> 1; j > 0; j >>= 1) {
      for (int i = tid; i < NPAD; i += 1024) {
        int ixj = i ^ j;
        if (ixj > i) {
          unsigned long long a = sk[i], bkey = sk[ixj];
          bool up = ((i & k) == 0);
          if ((a > bkey) == up) { sk[i] = bkey; sk[ixj] = a; }
        }
      }
      __syncthreads();
    }
  }

  for (int jj = tid; jj < NCAND; jj += 1024) {
    int src = (int)(sk[jj] & 0xFFFFFFFFu);
    out[OFF_IDS + jj]         = bid[src];
    out[OFF_BOX + 4 * jj + 0] = box[4 * src + 0];
    out[OFF_BOX + 4 * jj + 1] = box[4 * src + 1];
    out[OFF_BOX + 4 * jj + 2] = box[4 * src + 2];
    out[OFF_BOX + 4 * jj + 3] = box[4 * src + 3];
    out[OFF_LAB + jj]         = label[src];
    out[OFF_SCO + jj]         = score[src];
    valid_sorted[jj]          = validf[src];
  }
}

// ---------------------------------------------------------------------------
// Kernel 3: exact greedy NMS, single workgroup (32 waves), everything LDS-
// resident (~129 KB, only possible with CDNA5's 320KB WGP LDS). Sorted boxes
// staged with async global->LDS B128 copies. Since the sort puts all valid
// candidates in a prefix [0, nvalid), and invalid entries can neither keep
// nor suppress, the sequential greedy loop is bounded by nvalid (~2x fewer
// barriers on the dependency chain).
// ---------------------------------------------------------------------------
__global__ __launch_bounds__(1024) void yolo_nms(
    const float* __restrict__ out_ids, const float* __restrict__ out_box,
    const float* __restrict__ out_lab, const float* __restrict__ valid_sorted,
    float* __restrict__ out_keep) {
  __shared__ __align__(16) float sb[NCAND * 4];   // 98 KB sorted boxes
  __shared__ unsigned skeyz[NCAND];               // 24.5 KB packed (label<<6)|bid
  __shared__ unsigned char skeep[NCAND];          // 6.1 KB keep mask
  __shared__ int s_nvalid;
  int tid = threadIdx.x;

  if (tid == 0) s_nvalid = 0;
  __syncthreads();

  // Stage boxes via async tensor-path copies (ASYNCcnt), 16B per lane per op.
  for (int j = tid; j < NCAND; j += 1024)
    async_b128_g2l(out_box + 4 * j, &sb[4 * j]);

  int cnt = 0;
  for (int j = tid; j < NCAND; j += 1024) {
    int lab = (int)out_lab[j];   // [1..20]
    int ib  = (int)out_ids[j];   // [0..63]
    skeyz[j] = (unsigned)((lab << 6) | ib);
    int v = (valid_sorted[j] > 0.5f) ? 1 : 0;
    skeep[j] = (unsigned char)v;
    cnt += v;
  }
  atomicAdd(&s_nvalid, cnt);

  wait_async0();
  __syncthreads();
  int nvalid = s_nvalid;   // valid candidates are a sorted prefix

  // Exact greedy suppression: keep[j] &= ~(keep[i] & sup[i][j]) for i < j,
  // evaluated sequentially in sorted order (matches the fori_loop).
  for (int i = 0; i < nvalid; ++i) {
    __syncthreads();
    if (!skeep[i]) continue;           // uniform branch (same LDS value)
    float il = sb[4 * i + 0], it = sb[4 * i + 1];
    float ir = sb[4 * i + 2], ib2 = sb[4 * i + 3];
    float iarea = (ir - il) * (ib2 - it);
    unsigned ikey = skeyz[i];
    for (int j = i + 1 + tid; j < nvalid; j += 1024) {
      if (!skeep[j] || skeyz[j] != ikey) continue;
      float jl = sb[4 * j + 0], jt = sb[4 * j + 1];
      float jr = sb[4 * j + 2], jb = sb[4 * j + 3];
      float ltx = fmaxf(il, jl), lty = fmaxf(it, jt);
      float rbx = fminf(ir, jr), rby = fminf(ib2, jb);
      float w = fmaxf(rbx - ltx, 0.0f), h = fmaxf(rby - lty, 0.0f);
      float inter = w * h;
      float jarea = (jr - jl) * (jb - jt);
      float iou = inter / (iarea + jarea - inter + 1e-9f);
      if (iou > THR_NMS) skeep[j] = 0;
    }
  }
  __syncthreads();

  for (int j = tid; j < NCAND; j += 1024)
    out_keep[j] = skeep[j] ? 1.0f : 0.0f;
}

// ---------------------------------------------------------------------------
extern "C" void kernel_launch(void* const* d_in, const int* in_sizes, int n_in,
                              void* d_out, int out_size, void* d_ws, size_t ws_size,
                              hipStream_t stream) {
  (void)in_sizes; (void)n_in; (void)out_size; (void)ws_size;
  const float* pyolos = (const float*)d_in[0];
  float* out = (float*)d_out;

  // Workspace layout (bytes), ~277 KB total:
  char* ws = (char*)d_ws;
  unsigned long long* ws_key = (unsigned long long*)(ws);                  // N*8
  float* ws_box  = (float*)(ws + (size_t)NCAND * 8);                       // N*4*4
  float* ws_lab  = (float*)(ws + (size_t)NCAND * 8 + (size_t)NCAND * 16);
  float* ws_bid  = ws_lab + NCAND;
  float* ws_sco  = ws_bid + NCAND;
  float* ws_val  = ws_sco + NCAND;
  float* ws_valS = ws_val + NCAND;   // valid in sorted order

  yolo_decode<<<(CELLS + 255) / 256, 256, 0, stream>>>(
      pyolos, ws_key, ws_box, ws_lab, ws_bid, ws_sco, ws_val);

  yolo_sort<<<1, 1024, 0, stream>>>(
      ws_key, ws_box, ws_lab, ws_bid, ws_sco, ws_val, out, ws_valS);

  yolo_nms<<<1, 1024, 0, stream>>>(
      out + OFF_IDS, out + OFF_BOX, out + OFF_LAB, ws_valS, out + OFF_KEEP);
}